// GraphGRUODE_7172595384547
// MI455X (gfx1250) — compile-verified
//
#include <hip/hip_runtime.h>
#include <cstdint>
#include <cstddef>

typedef __attribute__((ext_vector_type(2))) float v2f;
typedef __attribute__((ext_vector_type(8))) float v8f;

#define HID 64

// ---------- helpers: order-preserving f32 <-> u32 key for atomic max ----------
__device__ __forceinline__ unsigned f32_key(float f) {
  unsigned u = __float_as_uint(f);
  return (u & 0x80000000u) ? ~u : (u | 0x80000000u);
}
__device__ __forceinline__ float key_f32(unsigned k) {
  unsigned u = (k & 0x80000000u) ? (k ^ 0x80000000u) : ~k;
  return __uint_as_float(u);
}

// ---------- 1) node / edge masks ----------
__global__ void mask_kernel(const float* __restrict__ t_p,
                            const float* __restrict__ node_t,
                            const float* __restrict__ edge_t,
                            const int* __restrict__ src, const int* __restrict__ dst,
                            unsigned char* __restrict__ nm, unsigned char* __restrict__ em,
                            int N, int E) {
  int i = blockIdx.x * blockDim.x + threadIdx.x;
  float t = *t_p;
  if (i < N) nm[i] = (node_t[i] >= t) ? 1 : 0;
  if (i < E) {
    int s = src[i], d = dst[i];
    em[i] = ((edge_t[i] <= t) && (node_t[s] >= t) && (node_t[d] >= t) && (s != d)) ? 1 : 0;
  }
}

// ---------- 2) fs = feat @ W  via V_WMMA_F32_16X16X4_F32 ----------
// One wave computes a full 16x64 output stripe (one M-tile x all 4 N-tiles).
// A fragments for all 16 K-steps are preloaded unconditionally (rows clamped:
// an OOB A-row only feeds a C-row that is never stored).
// W is staged in LDS with K-row pairs interleaved:
//   Wlds[p*128 + c*2 + q] = W[(2p+q)*HID + c]   (p = K/2, q = K&1, c = col)
// so each B fragment (W[ka][c], W[ka+1][c]) is ONE contiguous ds_load_b64
// landing directly in an even-aligned VGPR pair (no packing moves).
__global__ void gemm_wmma_kernel(const float* __restrict__ feat,
                                 const float* __restrict__ W,
                                 float* __restrict__ out, int N) {
  __shared__ float Wlds[HID * HID]; // 16 KB of 320 KB/WGP
  for (int idx = threadIdx.x; idx < HID * HID; idx += blockDim.x) {
    int p = idx >> 7, rem = idx & 127, c = rem >> 1, q = rem & 1;
    Wlds[idx] = W[(2 * p + q) * HID + c];
  }
  __syncthreads();

  int wave = (blockIdx.x * blockDim.x + threadIdx.x) >> 5;
  int lane = threadIdx.x & 31;
  int tilesM = (N + 15) >> 4;
  if (wave >= tilesM) return;          // whole-wave exit: EXEC all-1s for WMMA

  int m    = lane & 15;
  int half = lane >> 4;
  int row  = wave * 16 + m;
  int rowc = row < N ? row : N - 1;    // clamp instead of exec-guard
  const v2f* arow = (const v2f*)(feat + (size_t)rowc * HID);

  v2f a[16];
#pragma unroll
  for (int k = 0; k < 16; ++k) a[k] = arow[2 * k + half];  // clause of b64 loads

  const v2f* bl = (const v2f*)Wlds;    // v2f index: pair_row * 64 + col
  v8f c0 = {}, c1 = {}, c2 = {}, c3 = {};
#pragma unroll
  for (int k = 0; k < 16; ++k) {
    int p = 2 * k + half;              // K-pair row for this lane-half
    const v2f* br = bl + p * 64 + m;
    v2f b0 = br[0];
    v2f b1 = br[16];
    v2f b2 = br[32];
    v2f b3 = br[48];
    c0 = __builtin_amdgcn_wmma_f32_16x16x4_f32(false, a[k], false, b0, (short)0, c0, false, false);
    c1 = __builtin_amdgcn_wmma_f32_16x16x4_f32(false, a[k], false, b1, (short)0, c1, false, false);
    c2 = __builtin_amdgcn_wmma_f32_16x16x4_f32(false, a[k], false, b2, (short)0, c2, false, false);
    c3 = __builtin_amdgcn_wmma_f32_16x16x4_f32(false, a[k], false, b3, (short)0, c3, false, false);
  }
#pragma unroll
  for (int i = 0; i < 8; ++i) {
    int r = wave * 16 + i + 8 * half;
    if (r < N) {
      float* orow = out + (size_t)r * HID + m;
      orow[0]  = c0[i];
      orow[16] = c1[i];
      orow[32] = c2[i];
      orow[48] = c3[i];
    }
  }
}

// ---------- 3) el = fs@al, er = fs@ar : one wave32 per node ----------
__global__ void elr_kernel(const float* __restrict__ fs,
                           const float* __restrict__ al, const float* __restrict__ ar,
                           float* __restrict__ el, float* __restrict__ er, int N) {
  int gid  = blockIdx.x * blockDim.x + threadIdx.x;
  int n    = gid >> 5;
  int lane = gid & 31;
  if (n >= N) return;
  const float* f = fs + (size_t)n * HID;
  float v0 = f[lane], v1 = f[lane + 32];
  float sl = v0 * al[lane] + v1 * al[lane + 32];
  float sr = v0 * ar[lane] + v1 * ar[lane + 32];
#pragma unroll
  for (int off = 16; off > 0; off >>= 1) {
    sl += __shfl_down(sl, off, 32);
    sr += __shfl_down(sr, off, 32);
  }
  if (lane == 0) { el[n] = sl; er[n] = sr; }
}

// ---------- 4) segment max of leaky_relu(el[s]+er[d]) over dst ----------
__global__ void edge_max_kernel(const int* __restrict__ src, const int* __restrict__ dst,
                                const unsigned char* __restrict__ em,
                                const float* __restrict__ el, const float* __restrict__ er,
                                unsigned* __restrict__ mkey, int E) {
  int i = blockIdx.x * blockDim.x + threadIdx.x;
  if (i >= 2 * E) return;
  int e = (i < E) ? i : i - E;
  if (!em[e]) return;
  int s, d;
  if (i < E) { s = src[e]; d = dst[e]; } else { s = dst[e]; d = src[e]; }
  float x  = el[s] + er[d];
  float lr = x > 0.0f ? x : 0.2f * x;
  atomicMax(mkey + d, f32_key(lr));
}

// ---------- 5) accumulate Sum(w * fs[src]) and Sum(w): one wave32 per edge ----------
__global__ void edge_accum_kernel(const int* __restrict__ src, const int* __restrict__ dst,
                                  const unsigned char* __restrict__ em,
                                  const float* __restrict__ el, const float* __restrict__ er,
                                  const unsigned* __restrict__ mkey,
                                  const float* __restrict__ fs,
                                  float* __restrict__ num, float* __restrict__ den, int E) {
  int gid    = blockIdx.x * blockDim.x + threadIdx.x;
  int wid    = gid >> 5;
  int lane   = gid & 31;
  int nwaves = (gridDim.x * blockDim.x) >> 5;
  for (int i = wid; i < 2 * E; i += nwaves) {
    int e = (i < E) ? i : i - E;
    if (!em[e]) continue;
    int s, d;
    if (i < E) { s = src[e]; d = dst[e]; } else { s = dst[e]; d = src[e]; }
    float x  = el[s] + er[d];                 // uniform per wave (same addr -> broadcast)
    float lr = x > 0.0f ? x : 0.2f * x;
    float m  = key_f32(mkey[d]);
    float w  = __expf(lr - m);
    if (lane == 0) atomicAdd(den + d, w);
    const float* f = fs + (size_t)s * HID;
    atomicAdd(num + (size_t)d * HID + lane,      w * f[lane]);
    atomicAdd(num + (size_t)d * HID + lane + 32, w * f[lane + 32]);
  }
}

// ---------- 6) gat = num/den + bias ----------
__global__ void finalize_kernel(float* __restrict__ gat, const float* __restrict__ den,
                                const float* __restrict__ bias, int N) {
  int i = blockIdx.x * blockDim.x + threadIdx.x;
  if (i >= N * HID) return;
  int n = i >> 6, j = i & 63;
  float dn = den[n];
  float v  = (dn > 0.0f) ? gat[i] / dn : 0.0f;
  gat[i] = v + bias[j];
}

// ---------- 7) r = sigmoid(gat_xr + gat_hr); rh = r*h ----------
__global__ void r_rh_kernel(const float* __restrict__ gxr, const float* __restrict__ ghr,
                            const float* __restrict__ h, float* __restrict__ rh, int NH) {
  int i = blockIdx.x * blockDim.x + threadIdx.x;
  if (i >= NH) return;
  float r = 1.0f / (1.0f + __expf(-(gxr[i] + ghr[i])));
  rh[i] = r * h[i];
}

// ---------- 8) GRU gate + node-mask output ----------
__global__ void final_kernel(const float* __restrict__ gxz, const float* __restrict__ ghz,
                             const float* __restrict__ gxh, const float* __restrict__ ghh,
                             const float* __restrict__ h, const unsigned char* __restrict__ nm,
                             float* __restrict__ out, int N) {
  int i = blockIdx.x * blockDim.x + threadIdx.x;
  if (i >= N * HID) return;
  int n = i >> 6;
  float z  = 1.0f / (1.0f + __expf(-(gxz[i] + ghz[i])));
  float u  = tanhf(gxh[i] + ghh[i]);
  float dh = (1.0f - z) * (u - h[i]);
  out[i] = nm[n] ? dh : 0.0f;
}

extern "C" void kernel_launch(void* const* d_in, const int* in_sizes, int n_in,
                              void* d_out, int out_size, void* d_ws, size_t ws_size,
                              hipStream_t stream) {
  const float* t_p    = (const float*)d_in[0];
  const float* x      = (const float*)d_in[1];
  const float* h      = (const float*)d_in[2];
  const float* node_t = (const float*)d_in[3];
  const float* edge_t = (const float*)d_in[4];
  const int*   src    = (const int*)d_in[5];
  const int*   dst    = (const int*)d_in[6];
  int N = in_sizes[3];
  int E = in_sizes[4];

  // params flattened: for L in {xz,xr,xh,hz,hr,hh}: W, al, ar, b
  const float *Wp[6], *alp[6], *arp[6], *bp[6];
  for (int L = 0; L < 6; ++L) {
    Wp[L]  = (const float*)d_in[7 + 4 * L + 0];
    alp[L] = (const float*)d_in[7 + 4 * L + 1];
    arp[L] = (const float*)d_in[7 + 4 * L + 2];
    bp[L]  = (const float*)d_in[7 + 4 * L + 3];
  }

  // workspace carve-up (256B aligned)
  char* ws = (char*)d_ws;
  auto alloc = [&](size_t bytes) -> char* {
    char* p = ws; ws += (bytes + 255) & ~(size_t)255; return p;
  };
  float* fs = (float*)alloc((size_t)N * HID * 4);
  float* gat[6];
  for (int L = 0; L < 6; ++L) gat[L] = (float*)alloc((size_t)N * HID * 4);
  float* rh = (float*)alloc((size_t)N * HID * 4);
  float* el = (float*)alloc((size_t)N * 4);
  float* er = (float*)alloc((size_t)N * 4);
  unsigned* mkey = (unsigned*)alloc((size_t)N * 4);
  float* den = (float*)alloc((size_t)N * 4);
  unsigned char* nm = (unsigned char*)alloc((size_t)N);
  unsigned char* em = (unsigned char*)alloc((size_t)E);

  const int tpb = 256;
  int NE = (N > E) ? N : E;
  int gmask      = (NE + tpb - 1) / tpb;
  int tilesM     = (N + 15) / 16;               // one wave per 16x64 output stripe
  int gemmBlocks = (tilesM + 7) / 8;            // 8 waves per block
  int elrBlocks  = (N + 7) / 8;                 // 8 nodes per block (wave/node)
  int edgeBlocks = (2 * E + tpb - 1) / tpb;
  int accBlocks  = 2048;                        // persistent-style edge accumulation
  int nhBlocks   = (int)(((size_t)N * HID + tpb - 1) / tpb);

  mask_kernel<<<gmask, tpb, 0, stream>>>(t_p, node_t, edge_t, src, dst, nm, em, N, E);

  auto run_layer = [&](int L, const float* feat) {
    hipMemsetAsync(gat[L], 0, (size_t)N * HID * 4, stream);
    hipMemsetAsync(mkey, 0, (size_t)N * 4, stream);
    hipMemsetAsync(den, 0, (size_t)N * 4, stream);
    gemm_wmma_kernel<<<gemmBlocks, tpb, 0, stream>>>(feat, Wp[L], fs, N);
    elr_kernel<<<elrBlocks, tpb, 0, stream>>>(fs, alp[L], arp[L], el, er, N);
    edge_max_kernel<<<edgeBlocks, tpb, 0, stream>>>(src, dst, em, el, er, mkey, E);
    edge_accum_kernel<<<accBlocks, tpb, 0, stream>>>(src, dst, em, el, er, mkey, fs,
                                                     gat[L], den, E);
    finalize_kernel<<<nhBlocks, tpb, 0, stream>>>(gat[L], den, bp[L], N);
  };

  run_layer(0, x);   // xz
  run_layer(1, x);   // xr
  run_layer(2, x);   // xh
  run_layer(3, h);   // hz
  run_layer(4, h);   // hr
  r_rh_kernel<<<nhBlocks, tpb, 0, stream>>>(gat[1], gat[4], h, rh, N * HID);
  run_layer(5, rh);  // hh on r*h
  final_kernel<<<nhBlocks, tpb, 0, stream>>>(gat[0], gat[3], gat[2], gat[5], h, nm,
                                             (float*)d_out, N);
  (void)n_in; (void)out_size; (void)ws_size;
}